// generate_36575941492885
// MI455X (gfx1250) — compile-verified
//
#include <hip/hip_runtime.h>
#include <hip/hip_bf16.h>
#include <math.h>

// Problem constants
#define BB 32
#define SS 1024
#define DD 512
#define TWO_D 1024
#define THREE_D 1536
#define FIVE_D 2560
#define VV 50000
#define NVOC 49000
#define BN_EPS 1e-5f

typedef float v2f __attribute__((ext_vector_type(2)));
typedef float v4f __attribute__((ext_vector_type(4)));
typedef float v8f __attribute__((ext_vector_type(8)));

// Non-temporal (streaming) loads: data read exactly once -> TH_NT, keep L2 clean.
__device__ __forceinline__ v2f nt_load_v2f(const float* p) {
    return __builtin_nontemporal_load((const v2f*)p);
}
__device__ __forceinline__ v4f nt_load_v4f(const float* p) {
    return __builtin_nontemporal_load((const v4f*)p);
}
__device__ __forceinline__ float nt_load_f32(const float* p) {
    return __builtin_nontemporal_load(p);
}

// Exact tanh via hardware exp: tanh(x) = sign(x) * (1 - 2/(e^{2|x|}+1)).
// Overflow-safe: e^inf = inf -> 2/inf = 0 -> t = 1.
__device__ __forceinline__ float fast_tanhf(float x) {
    const float ax = fabsf(x);
    const float e  = __expf(2.0f * ax);
    const float t  = 1.0f - 2.0f / (e + 1.0f);
    return copysignf(t, x);
}

// ---------------------------------------------------------------------------
// Block reduction helpers (wave32)
// ---------------------------------------------------------------------------
__device__ __forceinline__ float wave_red_sum(float v) {
    for (int off = 16; off > 0; off >>= 1) v += __shfl_down(v, off, 32);
    return v;
}
__device__ __forceinline__ float wave_red_max(float v) {
    for (int off = 16; off > 0; off >>= 1) v = fmaxf(v, __shfl_down(v, off, 32));
    return v;
}
__device__ float block_red_sum(float v, float* sred) {
    __syncthreads();
    v = wave_red_sum(v);
    int lane = threadIdx.x & 31, w = threadIdx.x >> 5;
    int nw = blockDim.x >> 5;
    if (lane == 0) sred[w] = v;
    __syncthreads();
    if (w == 0) {
        float x = (threadIdx.x < (unsigned)nw) ? sred[threadIdx.x] : 0.0f;
        x = wave_red_sum(x);
        if (lane == 0) sred[0] = x;
    }
    __syncthreads();
    return sred[0];
}
__device__ float block_red_max(float v, float* sred) {
    __syncthreads();
    v = wave_red_max(v);
    int lane = threadIdx.x & 31, w = threadIdx.x >> 5;
    int nw = blockDim.x >> 5;
    if (lane == 0) sred[w] = v;
    __syncthreads();
    if (w == 0) {
        float x = (threadIdx.x < (unsigned)nw) ? sred[threadIdx.x] : -INFINITY;
        x = wave_red_max(x);
        if (lane == 0) sred[0] = x;
    }
    __syncthreads();
    return sred[0];
}

// ---------------------------------------------------------------------------
// WMMA f32 GEMM: C[32,N] = A[32,K] * W[N,K]^T + bias[N]
// One wave per 16-column tile, both 16-row M tiles (M=32 fixed), B fragment
// shared between the two WMMAs.  Software-pipelined: next K-step fragments
// are loaded before the current step's WMMAs consume theirs, so global loads
// overlap v_wmma execution instead of the serial load->wait->wmma pattern.
// W rows are touched by exactly one wave -> non-temporal loads.
// ---------------------------------------------------------------------------
__global__ __launch_bounds__(128)
void pg_gemm_wmma_f32(const float* __restrict__ A,
                      const float* __restrict__ W,
                      const float* __restrict__ bias,
                      float* __restrict__ C,
                      int N, int K)
{
    const int wave = threadIdx.x >> 5;
    const int lane = threadIdx.x & 31;
    const int tile_n = blockIdx.x * (blockDim.x >> 5) + wave;
    const int n0 = tile_n * 16;
    if (n0 >= N) return;                     // wave-uniform exit, EXEC stays full

    const int m    = lane & 15;              // A row / C column within tile
    const int half = lane >> 4;              // K-pair selector / C row-half
    const int n    = n0 + m;                 // output column owned by this lane
    const int nc   = (n < N) ? n : (N - 1);  // clamp for B loads on ragged tile
    const int h2   = half * 2;

    const float* __restrict__ arow0 = A + (size_t)m * K + h2;
    const float* __restrict__ arow1 = A + (size_t)(m + 16) * K + h2;
    const float* __restrict__ wrow  = W + (size_t)nc * K + h2;

    v8f c0 = {0.f,0.f,0.f,0.f,0.f,0.f,0.f,0.f};
    v8f c1 = {0.f,0.f,0.f,0.f,0.f,0.f,0.f,0.f};

    // Prologue: fragments for k = 0
    v2f a0 = *(const v2f*)(arow0);
    v2f a1 = *(const v2f*)(arow1);
    v2f b  = nt_load_v2f(wrow);

    #pragma unroll 8
    for (int k = 4; k < K; k += 4) {
        // Issue next step's loads first so they overlap the WMMAs below.
        v2f na0 = *(const v2f*)(arow0 + k);
        v2f na1 = *(const v2f*)(arow1 + k);
        v2f nb  = nt_load_v2f(wrow + k);
        c0 = __builtin_amdgcn_wmma_f32_16x16x4_f32(false, a0, false, b,
                                                   (short)0, c0, false, false);
        c1 = __builtin_amdgcn_wmma_f32_16x16x4_f32(false, a1, false, b,
                                                   (short)0, c1, false, false);
        a0 = na0; a1 = na1; b = nb;
    }
    // Epilogue
    c0 = __builtin_amdgcn_wmma_f32_16x16x4_f32(false, a0, false, b,
                                               (short)0, c0, false, false);
    c1 = __builtin_amdgcn_wmma_f32_16x16x4_f32(false, a1, false, b,
                                               (short)0, c1, false, false);

    if (n < N) {
        const float bv = bias ? bias[n] : 0.0f;
        #pragma unroll
        for (int r = 0; r < 8; ++r) {
            const int row = half * 8 + r;    // C rows: lanes0-15 -> 0..7, 16-31 -> 8..15
            C[(size_t)row        * N + n] = c0[r] + bv;
            C[(size_t)(row + 16) * N + n] = c1[r] + bv;
        }
    }
}

// ---------------------------------------------------------------------------
// Pack dec_op[b, 0:D]=h[0,b,:], dec_op[b, D:2D]=c[0,b,:]
// ---------------------------------------------------------------------------
__global__ void pg_pack_decop(const float* __restrict__ h,
                              const float* __restrict__ c,
                              float* __restrict__ dec_op)
{
    int i = blockIdx.x * blockDim.x + threadIdx.x;   // 0 .. B*2D-1
    if (i >= BB * TWO_D) return;
    int b = i / TWO_D, j = i % TWO_D;
    dec_op[i] = (j < DD) ? h[b * DD + j] : c[b * DD + (j - DD)];
}

// ---------------------------------------------------------------------------
// scores[b,s] = sum_d tanh(enc_op1[b,s,d] + dec_op1[b,d]) * w_attn[d]
// One wave per (b,s) row; enc_op1 is a 134 MB single-use stream -> NT loads.
// dec_op1/w_attn are heavily reused -> regular (cacheable) loads.
// ---------------------------------------------------------------------------
__global__ __launch_bounds__(256)
void pg_attn_scores(const float* __restrict__ enc_op1,
                    const float* __restrict__ dec_op1,
                    const float* __restrict__ w_attn,
                    float* __restrict__ scores)
{
    const int wave = threadIdx.x >> 5;
    const int lane = threadIdx.x & 31;
    const int row  = blockIdx.x * 8 + wave;          // 0 .. B*S-1
    const int b    = row >> 10;

    const float* __restrict__ e  = enc_op1 + (size_t)row * TWO_D;
    const float* __restrict__ dq = dec_op1 + (size_t)b * TWO_D;

    float acc = 0.0f;
    #pragma unroll
    for (int i = 0; i < 8; ++i) {
        const int d = i * 128 + lane * 4;
        v4f ev = nt_load_v4f(e + d);
        v4f dv = *(const v4f*)(dq + d);
        v4f wv = *(const v4f*)(w_attn + d);
        acc += fast_tanhf(ev.x + dv.x) * wv.x;
        acc += fast_tanhf(ev.y + dv.y) * wv.y;
        acc += fast_tanhf(ev.z + dv.z) * wv.z;
        acc += fast_tanhf(ev.w + dv.w) * wv.w;
    }
    acc = wave_red_sum(acc);
    if (lane == 0) scores[row] = acc;
}

// ---------------------------------------------------------------------------
// Masked softmax + renormalize.  attn = mask*exp(s - max_all) / sum(masked)
// (equals softmax -> *mask -> renorm exactly).  One block per batch row.
// ---------------------------------------------------------------------------
__global__ __launch_bounds__(256)
void pg_attn_softmax(const float* __restrict__ scores,
                     const int* __restrict__ src_inp,
                     float* __restrict__ attn_ws,
                     float* __restrict__ attn_out)
{
    __shared__ float sred[8];
    const int b = blockIdx.x, tid = threadIdx.x;
    const float* sc = scores + b * SS;

    float v[4];
    float lmax = -INFINITY;
    #pragma unroll
    for (int i = 0; i < 4; ++i) { v[i] = sc[tid + i * 256]; lmax = fmaxf(lmax, v[i]); }
    const float mx = block_red_max(lmax, sred);

    float e[4];
    float lsum = 0.0f;
    #pragma unroll
    for (int i = 0; i < 4; ++i) {
        const int s = tid + i * 256;
        const float msk = (src_inp[b * SS + s] != 0) ? 1.0f : 0.0f;
        e[i] = msk * __expf(v[i] - mx);
        lsum += e[i];
    }
    const float total = block_red_sum(lsum, sred);
    const float inv = 1.0f / total;
    #pragma unroll
    for (int i = 0; i < 4; ++i) {
        const int s = tid + i * 256;
        const float a = e[i] * inv;
        attn_ws[b * SS + s]  = a;
        attn_out[b * SS + s] = a;
    }
}

// ---------------------------------------------------------------------------
// context[b,d] = sum_s attn[b,s]*enc_op[b,s,d], then BatchNorm (eval mode).
// Grid (B, 2D/256); attn row cached in LDS; enc_op single-use -> NT loads.
// ---------------------------------------------------------------------------
__global__ __launch_bounds__(256)
void pg_context_bn(const float* __restrict__ enc_op,
                   const float* __restrict__ attn,
                   const float* __restrict__ gamma,
                   const float* __restrict__ beta,
                   const float* __restrict__ mean,
                   const float* __restrict__ var,
                   float* __restrict__ ctx_ws,
                   float* __restrict__ ctx_out)
{
    __shared__ float sa[SS];
    const int b = blockIdx.x;
    const int d = blockIdx.y * 256 + threadIdx.x;

    for (int i = threadIdx.x; i < SS; i += 256) sa[i] = attn[b * SS + i];
    __syncthreads();

    const float* __restrict__ ebase = enc_op + (size_t)b * SS * TWO_D + d;
    float acc = 0.0f;
    #pragma unroll 4
    for (int s = 0; s < SS; ++s)
        acc += sa[s] * nt_load_f32(ebase + (size_t)s * TWO_D);

    const float o = (acc - mean[d]) * rsqrtf(var[d] + BN_EPS) * gamma[d] + beta[d];
    ctx_ws[b * TWO_D + d]  = o;
    ctx_out[b * TWO_D + d] = o;
}

// ---------------------------------------------------------------------------
// Build temp[b,3D] = [dec_output ; ctx] and pg_in[b,5D] = [ctx ; dec_op ; dec_inp]
// Grid (B, 5D/256).
// ---------------------------------------------------------------------------
__global__ void pg_pack2(const float* __restrict__ dec_output,
                         const float* __restrict__ dec_inp,
                         const float* __restrict__ ctx,
                         const float* __restrict__ dec_op,
                         float* __restrict__ temp,
                         float* __restrict__ pg_in)
{
    const int b = blockIdx.x;
    const int j = blockIdx.y * 256 + threadIdx.x;    // 0 .. 5D-1

    // pg_in: [ctx(2D) ; dec_op(2D) ; dec_inp(D)]
    float pv;
    if (j < TWO_D)            pv = ctx[b * TWO_D + j];
    else if (j < 2 * TWO_D)   pv = dec_op[b * TWO_D + (j - TWO_D)];
    else                      pv = dec_inp[b * DD + (j - 2 * TWO_D)];
    pg_in[b * FIVE_D + j] = pv;

    // temp: [dec_output(D) ; ctx(2D)]
    if (j < THREE_D) {
        float tv = (j < DD) ? dec_output[b * DD + j] : ctx[b * TWO_D + (j - DD)];
        temp[b * THREE_D + j] = tv;
    }
}

// ---------------------------------------------------------------------------
// p_gen[b] = sigmoid(pg_in[b,:] . W_p + b_p).  One block per b.
// ---------------------------------------------------------------------------
__global__ __launch_bounds__(256)
void pg_pgen(const float* __restrict__ pg_in,
             const float* __restrict__ W_p,
             const float* __restrict__ b_p,
             float* __restrict__ pgen)
{
    __shared__ float sred[8];
    const int b = blockIdx.x, tid = threadIdx.x;
    float acc = 0.0f;
    #pragma unroll
    for (int i = 0; i < FIVE_D / 256; ++i) {
        const int j = tid + i * 256;
        acc += pg_in[b * FIVE_D + j] * W_p[j];
    }
    const float total = block_red_sum(acc, sred);
    if (tid == 0) {
        const float z = total + b_p[0];
        pgen[b] = 1.0f / (1.0f + __expf(-z));
    }
}

// ---------------------------------------------------------------------------
// p_final[b, 0:NVOC] = p_gen[b] * softmax(logits[b,:]); OOV tail = 0.
// One block (1024 threads) per b; 48 strided iterations cover 49000.
// logits are read 3x per row -> keep cacheable loads.
// ---------------------------------------------------------------------------
__global__ __launch_bounds__(1024)
void pg_vocab_softmax(const float* __restrict__ logits,
                      const float* __restrict__ pgen,
                      float* __restrict__ p_final)
{
    __shared__ float sred[32];
    const int b = blockIdx.x, tid = threadIdx.x;
    const float* lrow = logits + (size_t)b * NVOC;

    float lmax = -INFINITY;
    for (int i = 0; i < 48; ++i) {
        const int v = tid + i * 1024;
        if (v < NVOC) lmax = fmaxf(lmax, lrow[v]);
    }
    const float mx = block_red_max(lmax, sred);

    float lsum = 0.0f;
    for (int i = 0; i < 48; ++i) {
        const int v = tid + i * 1024;
        if (v < NVOC) lsum += __expf(lrow[v] - mx);
    }
    const float total = block_red_sum(lsum, sred);
    const float scale = pgen[b] / total;

    float* orow = p_final + (size_t)b * VV;
    for (int i = 0; i < 48; ++i) {
        const int v = tid + i * 1024;
        if (v < NVOC) orow[v] = scale * __expf(lrow[v] - mx);
    }
    if (tid < VV - NVOC) orow[NVOC + tid] = 0.0f;    // OOV padding
}

// ---------------------------------------------------------------------------
// Scatter: p_final[b, src_inp[b,s]] += (1 - p_gen[b]) * attn[b,s]
// ---------------------------------------------------------------------------
__global__ void pg_scatter(const int* __restrict__ src_inp,
                           const float* __restrict__ attn,
                           const float* __restrict__ pgen,
                           float* __restrict__ p_final)
{
    const int b = blockIdx.x;
    const int s = blockIdx.y * 256 + threadIdx.x;
    const int idx = src_inp[b * SS + s];
    const float val = (1.0f - pgen[b]) * attn[b * SS + s];
    atomicAdd(&p_final[(size_t)b * VV + idx], val);
}

// ---------------------------------------------------------------------------
// Epsilon: p += (p == 0) * 1e-12
// ---------------------------------------------------------------------------
__global__ void pg_epsilon(float* __restrict__ p_final)
{
    const int i = blockIdx.x * blockDim.x + threadIdx.x;
    if (i >= BB * VV) return;
    const float p = p_final[i];
    p_final[i] = p + ((p == 0.0f) ? 1e-12f : 0.0f);
}

// ---------------------------------------------------------------------------
extern "C" void kernel_launch(void* const* d_in, const int* in_sizes, int n_in,
                              void* d_out, int out_size, void* d_ws, size_t ws_size,
                              hipStream_t stream) {
    const float* dec_output = (const float*)d_in[0];   // [B,1,D]
    const float* dec_hid_h  = (const float*)d_in[1];   // [1,B,D]
    const float* dec_hid_c  = (const float*)d_in[2];   // [1,B,D]
    const float* dec_inp    = (const float*)d_in[3];   // [B,1,D]
    const float* enc_op     = (const float*)d_in[4];   // [B,S,2D]
    const float* enc_op1    = (const float*)d_in[5];   // [B,S,2D]
    const int*   src_inp    = (const int*)d_in[6];     // [B,S]
    const float* W_dec      = (const float*)d_in[7];   // [2D,2D]
    const float* b_dec      = (const float*)d_in[8];   // [2D]
    const float* w_attn     = (const float*)d_in[9];   // [2D]
    const float* W_v1       = (const float*)d_in[10];  // [D,3D]
    const float* b_v1       = (const float*)d_in[11];  // [D]
    const float* W_v2       = (const float*)d_in[12];  // [NVOC,D]
    const float* b_v2       = (const float*)d_in[13];  // [NVOC]
    const float* W_p        = (const float*)d_in[14];  // [1,5D]
    const float* b_p        = (const float*)d_in[15];  // [1]
    const float* bn_gamma   = (const float*)d_in[16];
    const float* bn_beta    = (const float*)d_in[17];
    const float* bn_mean    = (const float*)d_in[18];
    const float* bn_var     = (const float*)d_in[19];

    // Output layout: p_final [B,V] | attn [B,S] | context [B,2D]
    float* out_pfinal = (float*)d_out;
    float* out_attn   = out_pfinal + (size_t)BB * VV;
    float* out_ctx    = out_attn + (size_t)BB * SS;

    // Workspace layout (floats)
    float* ws = (float*)d_ws;
    float* ws_dec_op  = ws;                  ws += BB * TWO_D;     // 32768
    float* ws_dec_op1 = ws;                  ws += BB * TWO_D;     // 32768
    float* ws_scores  = ws;                  ws += BB * SS;        // 32768
    float* ws_attn    = ws;                  ws += BB * SS;        // 32768
    float* ws_ctx     = ws;                  ws += BB * TWO_D;     // 32768
    float* ws_temp    = ws;                  ws += BB * THREE_D;   // 49152
    float* ws_pgin    = ws;                  ws += BB * FIVE_D;    // 81920
    float* ws_hidden  = ws;                  ws += BB * DD;        // 16384
    float* ws_pgen    = ws;                  ws += BB;             // 32
    float* ws_logits  = ws;                  /* 32*49000 = 1568000 */

    // 1) dec_op = [h ; c]
    pg_pack_decop<<<(BB * TWO_D + 255) / 256, 256, 0, stream>>>(
        dec_hid_h, dec_hid_c, ws_dec_op);

    // 2) dec_op1 = dec_op @ W_dec^T + b_dec   (WMMA, N=1024, K=1024 -> 64 tiles)
    pg_gemm_wmma_f32<<<(TWO_D / 16 + 3) / 4, 128, 0, stream>>>(
        ws_dec_op, W_dec, b_dec, ws_dec_op1, TWO_D, TWO_D);

    // 3) attention scores (streams enc_op1, 134 MB, NT)
    pg_attn_scores<<<(BB * SS) / 8, 256, 0, stream>>>(
        enc_op1, ws_dec_op1, w_attn, ws_scores);

    // 4) masked softmax + renorm -> attn (also final output)
    pg_attn_softmax<<<BB, 256, 0, stream>>>(
        ws_scores, src_inp, ws_attn, out_attn);

    // 5) context + BatchNorm (streams enc_op, 134 MB, NT) -> context output
    pg_context_bn<<<dim3(BB, TWO_D / 256), 256, 0, stream>>>(
        enc_op, ws_attn, bn_gamma, bn_beta, bn_mean, bn_var, ws_ctx, out_ctx);

    // 6) pack temp [B,3D] and pg_in [B,5D]
    pg_pack2<<<dim3(BB, FIVE_D / 256), 256, 0, stream>>>(
        dec_output, dec_inp, ws_ctx, ws_dec_op, ws_temp, ws_pgin);

    // 7) hidden = temp @ W_v1^T + b_v1   (WMMA, N=512, K=1536 -> 32 tiles)
    pg_gemm_wmma_f32<<<(DD / 16 + 3) / 4, 128, 0, stream>>>(
        ws_temp, W_v1, b_v1, ws_hidden, DD, THREE_D);

    // 8) p_gen
    pg_pgen<<<BB, 256, 0, stream>>>(ws_pgin, W_p, b_p, ws_pgen);

    // 9) logits = hidden @ W_v2^T + b_v2 (WMMA, N=49000, K=512 -> 3063 tiles)
    {
        const int tiles = (NVOC + 15) / 16;          // 3063
        pg_gemm_wmma_f32<<<(tiles + 3) / 4, 128, 0, stream>>>(
            ws_hidden, W_v2, b_v2, ws_logits, NVOC, DD);
    }

    // 10) p_final = p_gen * softmax(logits) with OOV zero padding
    pg_vocab_softmax<<<BB, 1024, 0, stream>>>(ws_logits, ws_pgen, out_pfinal);

    // 11) scatter-add copy distribution
    pg_scatter<<<dim3(BB, SS / 256), 256, 0, stream>>>(
        src_inp, ws_attn, ws_pgen, out_pfinal);

    // 12) epsilon where zero
    pg_epsilon<<<(BB * VV + 255) / 256, 256, 0, stream>>>(out_pfinal);

    (void)in_sizes; (void)n_in; (void)out_size; (void)ws_size;
}